// GPTOSSMoELayer_77704548319529
// MI455X (gfx1250) — compile-verified
//
#include <hip/hip_runtime.h>
#include <hip/hip_bf16.h>

typedef __attribute__((ext_vector_type(8))) float v8f;
typedef __attribute__((ext_vector_type(2))) float v2f;

#define NEXP 32
#define TOPK 4
#define HID 1024
#define IMED 1024
#define NTOK 1024

static __device__ inline v8f wmma_f32_16x16x4(v2f a, v2f b, v8f c) {
    // D = A(16x4) * B(4x16) + C(16x16), fp32 WMMA
    return __builtin_amdgcn_wmma_f32_16x16x4_f32(
        /*neg_a=*/false, a, /*neg_b=*/false, b,
        /*c_mod=*/(short)0, c, /*reuse_a=*/false, /*reuse_b=*/false);
}

// Non-temporal v2f load: weights are single-use per call; keep them from
// displacing x/hbuf in the 192MB L2.
static __device__ inline v2f ld_nt(const float* p) {
    return __builtin_nontemporal_load((const v2f*)p);
}

// ---------------- Kernel 1: router (softmax + top-4, renormalized) ----------
// 1 wave per token; lane == expert (E == 32 == wave32).
__global__ void moe_router(const float* __restrict__ x,
                           const float* __restrict__ wg,
                           const float* __restrict__ bg,
                           int* __restrict__ topi,
                           float* __restrict__ topw) {
    const int t = blockIdx.x;
    const int lane = threadIdx.x;

    const float* xr = x + (size_t)t * HID;
    const float* wr = wg + (size_t)lane * HID;
    float acc = 0.f;
#pragma unroll 4
    for (int j = 0; j < HID; ++j) acc += xr[j] * wr[j];
    acc += bg[lane];

    // wave softmax
    float m = acc;
    for (int o = 16; o > 0; o >>= 1) m = fmaxf(m, __shfl_xor(m, o, 32));
    float p = __expf(acc - m);
    float s = p;
    for (int o = 16; o > 0; o >>= 1) s += __shfl_xor(s, o, 32);
    float prob = p / s;

    // iterative top-4 (ties -> lowest lane, deterministic)
    float v = prob;
    int sel_i[TOPK];
    float sel_v[TOPK];
    for (int k = 0; k < TOPK; ++k) {
        float mv = v;
        for (int o = 16; o > 0; o >>= 1) mv = fmaxf(mv, __shfl_xor(mv, o, 32));
        unsigned long long bal = __ballot(v == mv);
        int sl = __ffsll(bal) - 1;
        sel_i[k] = sl;
        sel_v[k] = mv;
        if (lane == sl) v = -1.0f;
    }
    if (lane == 0) {
        float ssum = sel_v[0] + sel_v[1] + sel_v[2] + sel_v[3];
        float inv = 1.0f / ssum;
#pragma unroll
        for (int k = 0; k < TOPK; ++k) {
            topi[t * TOPK + k] = sel_i[k];
            topw[t * TOPK + k] = sel_v[k] * inv;
        }
    }
}

// ---------------- Kernel 2: deterministic per-expert token lists ------------
// 1 wave per expert; tokens appended in ascending token order via ballot scan.
__global__ void moe_build_lists(const int* __restrict__ topi,
                                const float* __restrict__ topw,
                                int* __restrict__ tok_list,
                                float* __restrict__ wt_list,
                                int* __restrict__ count) {
    const int e = blockIdx.x;
    const int lane = threadIdx.x;
    int base = 0;
    for (int t0 = 0; t0 < NTOK; t0 += 32) {
        const int t = t0 + lane;
        int found = 0;
        float w = 0.f;
#pragma unroll
        for (int k = 0; k < TOPK; ++k) {
            if (topi[t * TOPK + k] == e) { found = 1; w = topw[t * TOPK + k]; }
        }
        unsigned mask = (unsigned)__ballot(found);
        int pos = base + __popc(mask & ((1u << lane) - 1u));
        if (found) {
            tok_list[e * NTOK + pos] = t;
            wt_list[e * NTOK + pos] = w;
        }
        base += __popc(mask);
    }
    if (lane == 0) count[e] = base;
}

// ---------------- Kernel 3: exclusive prefix over counts --------------------
__global__ void moe_prefix(const int* __restrict__ count, int* __restrict__ offs) {
    if (threadIdx.x == 0 && blockIdx.x == 0) {
        int a = 0;
        for (int e = 0; e < NEXP; ++e) { offs[e] = a; a += count[e]; }
        offs[NEXP] = a;
    }
}

// ---------------- Kernel 4: gate+up proj + SwiGLU (WMMA f32) ----------------
// grid(E, I/16); 8 waves/block; each wave owns one 16-token x 16-channel tile.
__global__ void moe_gateup(const float* __restrict__ x,
                           const float* __restrict__ w1,
                           const float* __restrict__ b1,
                           const int* __restrict__ tok_list,
                           const int* __restrict__ count,
                           const int* __restrict__ offs,
                           float* __restrict__ hbuf) {
    const int e = blockIdx.x;
    const int n0 = blockIdx.y * 16;
    const int cnt = count[e];
    if (cnt == 0) return;

    const int wave = threadIdx.x >> 5;
    const int lane = threadIdx.x & 31;
    const int nl = lane & 15;          // N (B/D column) and M (A row) sub-index
    const int hi = lane >> 4;          // half-wave -> K-offset / M-offset
    const int koff = hi * 2;

    const float* bgrow = w1 + ((size_t)e * (2 * IMED) + (n0 + nl)) * HID;
    const float* burow = bgrow + (size_t)IMED * HID;
    const float biasg = b1[e * (2 * IMED) + n0 + nl];
    const float biasu = b1[e * (2 * IMED) + IMED + n0 + nl];
    const int rowoff = offs[e];

    const int mtiles = (cnt + 15) >> 4;
    for (int mt = wave; mt < mtiles; mt += 8) {
        const int m0 = mt * 16;
        const int mrow = min(m0 + nl, cnt - 1);          // pad with last valid
        const int trow = tok_list[e * NTOK + mrow];
        const float* xrow = x + (size_t)trow * HID;

        v8f g = {};
        v8f u = {};
#pragma unroll 4
        for (int k = 0; k < HID; k += 4) {
            v2f a  = *(const v2f*)(xrow + k + koff);   // x is reused: keep RT
            v2f bg = ld_nt(bgrow + k + koff);          // weights: NT stream
            v2f bu = ld_nt(burow + k + koff);
            g = wmma_f32_16x16x4(a, bg, g);
            u = wmma_f32_16x16x4(a, bu, u);
        }

        const int rowbase = rowoff + m0;
#pragma unroll
        for (int r = 0; r < 8; ++r) {
            const int M = r + hi * 8;
            if (m0 + M < cnt) {
                float gg = g[r] + biasg;
                float uu = u[r] + biasu;
                // SiLU(g)*u with fast reciprocal (v_rcp_f32)
                float sig = __builtin_amdgcn_rcpf(1.0f + __expf(-gg));
                hbuf[(size_t)(rowbase + M) * IMED + n0 + nl] = gg * sig * uu;
            }
        }
    }
}

// ---------------- Kernel 5: down proj + weighted scatter-add ----------------
// grid(E, H/16); atomicAdd f32 into zeroed output (4 adds per element total).
__global__ void moe_down(const float* __restrict__ w2,
                         const float* __restrict__ b2,
                         const int* __restrict__ tok_list,
                         const float* __restrict__ wt_list,
                         const int* __restrict__ count,
                         const int* __restrict__ offs,
                         const float* __restrict__ hbuf,
                         float* __restrict__ out) {
    const int e = blockIdx.x;
    const int n0 = blockIdx.y * 16;
    const int cnt = count[e];
    if (cnt == 0) return;

    const int wave = threadIdx.x >> 5;
    const int lane = threadIdx.x & 31;
    const int nl = lane & 15;
    const int hi = lane >> 4;
    const int koff = hi * 2;

    const float* brow = w2 + ((size_t)e * HID + n0 + nl) * IMED;
    const float bias = b2[e * HID + n0 + nl];
    const int rowoff = offs[e];

    const int mtiles = (cnt + 15) >> 4;
    for (int mt = wave; mt < mtiles; mt += 8) {
        const int m0 = mt * 16;
        const int arow = rowoff + min(m0 + nl, cnt - 1);
        const float* hrow = hbuf + (size_t)arow * IMED;

        v8f y = {};
#pragma unroll 4
        for (int k = 0; k < IMED; k += 4) {
            v2f a = *(const v2f*)(hrow + k + koff);   // hbuf reused: RT
            v2f b = ld_nt(brow + k + koff);           // weights: NT stream
            y = wmma_f32_16x16x4(a, b, y);
        }

#pragma unroll
        for (int r = 0; r < 8; ++r) {
            const int M = r + hi * 8;
            const int mm = m0 + M;
            if (mm < cnt) {
                const int tok = tok_list[e * NTOK + mm];
                const float c = wt_list[e * NTOK + mm];
                atomicAdd(&out[(size_t)tok * HID + n0 + nl], c * (y[r] + bias));
            }
        }
    }
}

extern "C" void kernel_launch(void* const* d_in, const int* in_sizes, int n_in,
                              void* d_out, int out_size, void* d_ws, size_t ws_size,
                              hipStream_t stream) {
    const float* x  = (const float*)d_in[0];   // [T,H]
    const float* wg = (const float*)d_in[1];   // [E,H]
    const float* bg = (const float*)d_in[2];   // [E]
    const float* w1 = (const float*)d_in[3];   // [E,2I,H]
    const float* b1 = (const float*)d_in[4];   // [E,2I]
    const float* w2 = (const float*)d_in[5];   // [E,H,I]
    const float* b2 = (const float*)d_in[6];   // [E,H]
    float* out = (float*)d_out;                // [T,H]

    // workspace carve-up (256B aligned slabs)
    char* p = (char*)d_ws;
    auto carve = [&](size_t bytes) {
        char* r = p;
        p += (bytes + 255) & ~(size_t)255;
        return (void*)r;
    };
    int*   topi     = (int*)  carve((size_t)NTOK * TOPK * sizeof(int));
    float* topw     = (float*)carve((size_t)NTOK * TOPK * sizeof(float));
    int*   count    = (int*)  carve((size_t)NEXP * sizeof(int));
    int*   offs     = (int*)  carve((size_t)(NEXP + 1) * sizeof(int));
    int*   tok_list = (int*)  carve((size_t)NEXP * NTOK * sizeof(int));
    float* wt_list  = (float*)carve((size_t)NEXP * NTOK * sizeof(float));
    float* hbuf     = (float*)carve((size_t)NTOK * TOPK * IMED * sizeof(float));

    hipMemsetAsync(out, 0, (size_t)NTOK * HID * sizeof(float), stream);

    moe_router<<<NTOK, 32, 0, stream>>>(x, wg, bg, topi, topw);
    moe_build_lists<<<NEXP, 32, 0, stream>>>(topi, topw, tok_list, wt_list, count);
    moe_prefix<<<1, 1, 0, stream>>>(count, offs);
    moe_gateup<<<dim3(NEXP, IMED / 16), 256, 0, stream>>>(x, w1, b1, tok_list, count, offs, hbuf);
    moe_down<<<dim3(NEXP, HID / 16), 256, 0, stream>>>(w2, b2, tok_list, wt_list, count, offs, hbuf, out);
}